// PrototypeAlignmentLoss_64321430225252
// MI455X (gfx1250) — compile-verified
//
#include <hip/hip_runtime.h>

// -------- problem constants --------
#define DDIM       256
#define KPROTO     8
#define TILE_M     16
#define NBLK       512            // partial-sum blocks (deterministic 2-stage reduce)
#define LDS_STRIDE 260            // 256 + 4-float pad -> spreads rows across LDS banks
#define MOM        0.9f
#define NORM_EPS   1e-12f

typedef __attribute__((ext_vector_type(2))) float v2f;
typedef __attribute__((ext_vector_type(8))) float v8f;
typedef __attribute__((__vector_size__(16))) int  b128_t;   // matches builtin param

// ---- CDNA5 async global->LDS path (ASYNCcnt-tracked, VGPR-bypassing) ----
#if defined(__gfx1250__) && __has_builtin(__builtin_amdgcn_global_load_async_to_lds_b128)
#define ASYNC_LDS 1
#else
#define ASYNC_LDS 0
#endif

#if ASYNC_LDS
typedef __attribute__((address_space(1))) b128_t* gptr_b128;
typedef __attribute__((address_space(3))) b128_t* lptr_b128;
#endif

// Prefetch one 16x256 f32 tile (16KB) into an LDS buffer: 32 x b128 per wave.
__device__ __forceinline__ void tile_prefetch(const float* __restrict__ g,
                                              float* lds, int lane) {
#if ASYNC_LDS
    #pragma unroll
    for (int i = 0; i < 32; ++i) {
        int idx = i * 32 + lane;          // float4 index within tile
        int r   = idx >> 6;               // 64 float4 per row
        int c4  = idx & 63;
        __builtin_amdgcn_global_load_async_to_lds_b128(
            (gptr_b128)(const_cast<float*>(g) + idx * 4),
            (lptr_b128)&lds[r * LDS_STRIDE + c4 * 4],
            0, 0);
    }
#else
    const float4* g4 = (const float4*)g;
    #pragma unroll
    for (int i = 0; i < 32; ++i) {
        int idx = i * 32 + lane;
        int r   = idx >> 6;
        int c4  = idx & 63;
        *(float4*)&lds[r * LDS_STRIDE + c4 * 4] = g4[idx];
    }
#endif
}

// wait until <=32 async ops outstanding (previous tile complete, next in flight)
__device__ __forceinline__ void wait_async_tail() {
#if ASYNC_LDS
 #if __has_builtin(__builtin_amdgcn_s_wait_asynccnt)
    __builtin_amdgcn_s_wait_asynccnt(32);
 #else
    asm volatile("s_wait_asynccnt 0x20" ::: "memory");
 #endif
#endif
}
__device__ __forceinline__ void wait_async_all() {
#if ASYNC_LDS
 #if __has_builtin(__builtin_amdgcn_s_wait_asynccnt)
    __builtin_amdgcn_s_wait_asynccnt(0);
 #else
    asm volatile("s_wait_asynccnt 0x0" ::: "memory");
 #endif
#endif
}

// WMMA inner product of one 16-row tile against the (zero-padded 16-row)
// prototype tile. Each lane pulls a float4 (one ds_load_b128) per operand
// feeding TWO 16x16x4 WMMA steps; same K-permutation on A and B => exact dot.
// Also returns this lane's half-row sum-of-squares partial in 'ssq'.
__device__ __forceinline__ v8f tile_wmma(const float* aRow, const float* bRow,
                                         int koff, float& ssq) {
    v8f acc = {};
    float s = 0.0f;
    #pragma unroll 4
    for (int j = 0; j < DDIM / 8; ++j) {
        float4 a4 = *(const float4*)&aRow[8 * j + koff];
        float4 b4 = *(const float4*)&bRow[8 * j + koff];
        s += a4.x * a4.x + a4.y * a4.y + a4.z * a4.z + a4.w * a4.w;
        v2f a0 = {a4.x, a4.y}, b0 = {b4.x, b4.y};
        acc = __builtin_amdgcn_wmma_f32_16x16x4_f32(
                  false, a0, false, b0, (short)0, acc, false, false);
        v2f a1 = {a4.z, a4.w}, b1 = {b4.z, b4.w};
        acc = __builtin_amdgcn_wmma_f32_16x16x4_f32(
                  false, a1, false, b1, (short)0, acc, false, false);
    }
    ssq = s;
    return acc;
}

// =====================================================================
// Kernel A: source pass — WMMA dots vs prototypes, per-row argmax,
// scatter-accumulate normalized rows into per-block cluster partials.
// One wave per block, async-LDS double-buffered -> deterministic.
// =====================================================================
__global__ __launch_bounds__(32) void src_assign_kernel(
    const float* __restrict__ src, const float* __restrict__ protos,
    float* __restrict__ accumOut, float* __restrict__ countOut, int nRows)
{
    __shared__ __align__(16) float sSrcA[TILE_M * LDS_STRIDE];
    __shared__ __align__(16) float sSrcB[TILE_M * LDS_STRIDE];
    __shared__ __align__(16) float sProto[16 * LDS_STRIDE];   // rows 8..15 zero
    __shared__ __align__(16) float sAccum[KPROTO * DDIM];
    __shared__ float sCount[KPROTO];
    __shared__ float sSims[TILE_M * KPROTO];
    __shared__ float sInv[TILE_M];
    __shared__ int   sAssign[TILE_M];

    const int lane = threadIdx.x;
    const int blk  = blockIdx.x;

    const int nTiles      = nRows / TILE_M;
    const int tilesPerBlk = (nTiles + NBLK - 1) / NBLK;
    const int tileBase    = blk * tilesPerBlk;
    int tcount = nTiles - tileBase;
    if (tcount > tilesPerBlk) tcount = tilesPerBlk;
    if (tcount < 0) tcount = 0;

    if (tcount > 0)  // kick off first tile before doing LDS setup
        tile_prefetch(src + (size_t)tileBase * TILE_M * DDIM, sSrcA, lane);

    // prototypes -> LDS rows 0..7, zeros in rows 8..15 (unconditional B loads)
    for (int i = lane; i < 16 * DDIM; i += 32) {
        int r = i / DDIM, c = i % DDIM;
        sProto[r * LDS_STRIDE + c] = (r < KPROTO) ? protos[r * DDIM + c] : 0.0f;
    }
    for (int i = lane; i < KPROTO * DDIM; i += 32) sAccum[i] = 0.0f;
    if (lane < KPROTO) sCount[lane] = 0.0f;
    __syncthreads();

    const int mrow = lane & 15;
    const int koff = (lane >> 4) * 4;     // which K-quartet this half-wave feeds
    const float* aRowA = &sSrcA[mrow * LDS_STRIDE];
    const float* aRowB = &sSrcB[mrow * LDS_STRIDE];
    const float* bRow  = &sProto[mrow * LDS_STRIDE];

    for (int t = 0; t < tcount; ++t) {
        if (t + 1 < tcount) {
            tile_prefetch(src + (size_t)(tileBase + t + 1) * TILE_M * DDIM,
                          (t & 1) ? sSrcA : sSrcB, lane);
            wait_async_tail();
        } else {
            wait_async_all();
        }
        __syncthreads();

        const float* aRow = (t & 1) ? aRowB : aRowA;
        const float* bufc = (t & 1) ? sSrcB : sSrcA;

        float ssq;
        v8f acc = tile_wmma(aRow, bRow, koff, ssq);

        // combine the two K-halves of the row sumsq
        ssq += __shfl_xor(ssq, 16, 32);
        if (lane < 16)
            sInv[lane] = 1.0f / fmaxf(sqrtf(ssq), NORM_EPS);

        // C layout: lane = N column; VGPR r -> row r (lanes 0-15) / 8+r (16-31)
        {
            int ncol  = lane & 15;
            int mbase = (lane >> 4) * 8;
            if (ncol < KPROTO) {
                #pragma unroll
                for (int r = 0; r < 8; ++r)
                    sSims[(mbase + r) * KPROTO + ncol] = acc[r];
            }
        }
        __syncthreads();

        // first-max argmax per row (matches jnp.argmax tie-breaking)
        if (lane < 16) {
            const float* s = &sSims[lane * KPROTO];
            float best = s[0]; int bi = 0;
            #pragma unroll
            for (int k = 1; k < KPROTO; ++k)
                if (s[k] > best) { best = s[k]; bi = k; }
            sAssign[lane] = bi;
        }
        __syncthreads();

        // deterministic scatter, float4-wide: rows sequential, lanes disjoint
        for (int r = 0; r < TILE_M; ++r) {
            int   c   = sAssign[r];
            float inv = sInv[r];
            #pragma unroll
            for (int i = 0; i < DDIM / 128; ++i) {      // 2 iters
                int d = (lane + 32 * i) * 4;
                float4 sv = *(const float4*)&bufc[r * LDS_STRIDE + d];
                float4 av = *(float4*)&sAccum[c * DDIM + d];
                av.x += sv.x * inv; av.y += sv.y * inv;
                av.z += sv.z * inv; av.w += sv.w * inv;
                *(float4*)&sAccum[c * DDIM + d] = av;
            }
            if (lane == 0) sCount[c] += 1.0f;
        }
        __syncthreads();
    }

    float* outA = accumOut + (size_t)blk * (KPROTO * DDIM);
    for (int i = lane; i < KPROTO * DDIM; i += 32) outA[i] = sAccum[i];
    if (lane < KPROTO) countOut[blk * KPROTO + lane] = sCount[lane];
}

// =====================================================================
// Kernel B: fixed-order reduce of NBLK partials, momentum update +
// double normalization. One block; wave w owns cluster w.
// =====================================================================
__global__ __launch_bounds__(256) void proto_update_kernel(
    const float* __restrict__ protos, const float* __restrict__ accumIn,
    const float* __restrict__ countIn, float* __restrict__ protosNew)
{
    __shared__ float sSum[KPROTO * DDIM];
    __shared__ float sCnt[KPROTO];
    const int t = threadIdx.x;

    #pragma unroll
    for (int i = 0; i < (KPROTO * DDIM) / 256; ++i) {
        int e = t + 256 * i;
        float s = 0.0f;
        for (int b = 0; b < NBLK; ++b)           // ascending: deterministic
            s += accumIn[(size_t)b * (KPROTO * DDIM) + e];
        sSum[e] = s;
    }
    if (t < KPROTO) {
        float c = 0.0f;
        for (int b = 0; b < NBLK; ++b) c += countIn[b * KPROTO + t];
        sCnt[t] = c;
    }
    __syncthreads();

    const int wave = t >> 5;     // cluster id 0..7
    const int lane = t & 31;
    const float cnt = sCnt[wave];

    float mvals[DDIM / 32];
    float ssq = 0.0f;
    #pragma unroll
    for (int i = 0; i < DDIM / 32; ++i) {
        float m = sSum[wave * DDIM + lane + 32 * i] / fmaxf(cnt, 1.0f);
        mvals[i] = m;
        ssq += m * m;
    }
    for (int off = 16; off >= 1; off >>= 1) ssq += __shfl_xor(ssq, off, 32);
    const float invm = 1.0f / fmaxf(sqrtf(ssq), NORM_EPS);

    float fvals[DDIM / 32];
    float fssq = 0.0f;
    #pragma unroll
    for (int i = 0; i < DDIM / 32; ++i) {
        float p = protos[wave * DDIM + lane + 32 * i];
        float f = (cnt > 0.0f) ? (MOM * p + (1.0f - MOM) * mvals[i] * invm) : p;
        fvals[i] = f;
        fssq += f * f;
    }
    for (int off = 16; off >= 1; off >>= 1) fssq += __shfl_xor(fssq, off, 32);
    const float invf = 1.0f / fmaxf(sqrtf(fssq), NORM_EPS);

    #pragma unroll
    for (int i = 0; i < DDIM / 32; ++i)
        protosNew[wave * DDIM + lane + 32 * i] = fvals[i] * invf;
}

// =====================================================================
// Kernel C: target pass — WMMA dots vs new prototypes,
// loss = 1 - max_k(dot_k)/||t||; per-block deterministic partial sum.
// =====================================================================
__global__ __launch_bounds__(32) void target_loss_kernel(
    const float* __restrict__ tgt, const float* __restrict__ protosNew,
    float* __restrict__ lossPartials, int nRows)
{
    __shared__ __align__(16) float sSrcA[TILE_M * LDS_STRIDE];
    __shared__ __align__(16) float sSrcB[TILE_M * LDS_STRIDE];
    __shared__ __align__(16) float sProto[16 * LDS_STRIDE];
    __shared__ float sSims[TILE_M * KPROTO];

    const int lane = threadIdx.x;
    const int blk  = blockIdx.x;

    const int nTiles      = nRows / TILE_M;
    const int tilesPerBlk = (nTiles + NBLK - 1) / NBLK;
    const int tileBase    = blk * tilesPerBlk;
    int tcount = nTiles - tileBase;
    if (tcount > tilesPerBlk) tcount = tilesPerBlk;
    if (tcount < 0) tcount = 0;

    if (tcount > 0)
        tile_prefetch(tgt + (size_t)tileBase * TILE_M * DDIM, sSrcA, lane);

    for (int i = lane; i < 16 * DDIM; i += 32) {
        int r = i / DDIM, c = i % DDIM;
        sProto[r * LDS_STRIDE + c] = (r < KPROTO) ? protosNew[r * DDIM + c] : 0.0f;
    }
    __syncthreads();

    const int mrow = lane & 15;
    const int koff = (lane >> 4) * 4;
    const float* aRowA = &sSrcA[mrow * LDS_STRIDE];
    const float* aRowB = &sSrcB[mrow * LDS_STRIDE];
    const float* bRow  = &sProto[mrow * LDS_STRIDE];

    float lsum = 0.0f;

    for (int t = 0; t < tcount; ++t) {
        if (t + 1 < tcount) {
            tile_prefetch(tgt + (size_t)(tileBase + t + 1) * TILE_M * DDIM,
                          (t & 1) ? sSrcA : sSrcB, lane);
            wait_async_tail();
        } else {
            wait_async_all();
        }
        __syncthreads();

        const float* aRow = (t & 1) ? aRowB : aRowA;

        float ssq;
        v8f acc = tile_wmma(aRow, bRow, koff, ssq);
        ssq += __shfl_xor(ssq, 16, 32);

        {
            int ncol  = lane & 15;
            int mbase = (lane >> 4) * 8;
            if (ncol < KPROTO) {
                #pragma unroll
                for (int r = 0; r < 8; ++r)
                    sSims[(mbase + r) * KPROTO + ncol] = acc[r];
            }
        }
        __syncthreads();

        if (lane < 16) {
            const float* s = &sSims[lane * KPROTO];
            float best = s[0];
            #pragma unroll
            for (int k = 1; k < KPROTO; ++k) best = fmaxf(best, s[k]);
            float n = fmaxf(sqrtf(ssq), NORM_EPS);
            lsum += 1.0f - best / n;
        }
        __syncthreads();
    }

    for (int off = 16; off >= 1; off >>= 1) lsum += __shfl_xor(lsum, off, 32);
    if (lane == 0) lossPartials[blk] = lsum;
}

// =====================================================================
// Kernel D: fixed-order reduce of loss partials, write mean.
// =====================================================================
__global__ __launch_bounds__(32) void finalize_kernel(
    const float* __restrict__ lossPartials, float* __restrict__ out, int nRows)
{
    const int lane = threadIdx.x;
    float s = 0.0f;
    for (int i = lane; i < NBLK; i += 32) s += lossPartials[i];
    for (int off = 16; off >= 1; off >>= 1) s += __shfl_xor(s, off, 32);
    if (lane == 0) out[0] = s / (float)nRows;
}

// =====================================================================
extern "C" void kernel_launch(void* const* d_in, const int* in_sizes, int n_in,
                              void* d_out, int out_size, void* d_ws, size_t ws_size,
                              hipStream_t stream) {
    const float* src    = (const float*)d_in[0];
    const float* tgt    = (const float*)d_in[1];
    const float* protos = (const float*)d_in[2];

    const int N  = in_sizes[0] / DDIM;   // 131072
    const int Nt = in_sizes[1] / DDIM;   // 131072

    // workspace layout (floats)
    float* ws        = (float*)d_ws;
    float* wsAccum   = ws;                                      // NBLK * 8 * 256
    float* wsCount   = wsAccum + (size_t)NBLK * KPROTO * DDIM;  // NBLK * 8
    float* wsProtoNu = wsCount + (size_t)NBLK * KPROTO;         // 8 * 256
    float* wsLoss    = wsProtoNu + KPROTO * DDIM;               // NBLK
    (void)ws_size; (void)n_in; (void)out_size;

    src_assign_kernel<<<NBLK, 32, 0, stream>>>(src, protos, wsAccum, wsCount, N);
    proto_update_kernel<<<1, 256, 0, stream>>>(protos, wsAccum, wsCount, wsProtoNu);
    target_loss_kernel<<<NBLK, 32, 0, stream>>>(tgt, wsProtoNu, wsLoss, Nt);
    finalize_kernel<<<1, 32, 0, stream>>>(wsLoss, (float*)d_out, Nt);
}